// GraphProcessor_90065464197709
// MI455X (gfx1250) — compile-verified
//
#include <hip/hip_runtime.h>

#define NN 100000
#define NE 1600000
#define NG 64
#define BN_EPS 1e-5f

typedef __attribute__((ext_vector_type(2))) float v2f;
typedef __attribute__((ext_vector_type(8))) float v8f;

// ---------------------------------------------------------------- degrees
__global__ void deg_count_k(const int* __restrict__ dst, float* __restrict__ deg) {
    int e = blockIdx.x * 256 + threadIdx.x;
    if (e < NE) atomicAdd(&deg[dst[e]], 1.0f);
}

__global__ void deg_invert_k(float* __restrict__ deg) {
    int i = blockIdx.x * 256 + threadIdx.x;
    if (i < NN) deg[i] = 1.0f / fmaxf(deg[i], 1.0f);
}

// ---------------------------------------------------------------- layer 0 (din=7)
__global__ void scatter7_k(const int* __restrict__ src, const int* __restrict__ dst,
                           const float* __restrict__ x, float* __restrict__ agg) {
    int e = blockIdx.x * 256 + threadIdx.x;
    if (e < NE) {
        int s = src[e], d = dst[e];
#pragma unroll
        for (int k = 0; k < 7; ++k)
            atomicAdd(&agg[d * 7 + k], x[s * 7 + k]);
    }
}

__global__ __launch_bounds__(128)
void layer0_k(const float* __restrict__ x, const float* __restrict__ agg,
              const float* __restrict__ inv,
              const float* __restrict__ Wl, const float* __restrict__ bl,
              const float* __restrict__ Wr,
              const float* __restrict__ g,  const float* __restrict__ be,
              const float* __restrict__ rm, const float* __restrict__ rv,
              float* __restrict__ out) {
    int n = blockIdx.x;      // node
    int c = threadIdx.x;     // output column (0..127)
    float iv  = inv[n];
    float acc = bl[c];
#pragma unroll
    for (int k = 0; k < 7; ++k) {
        acc += x[n * 7 + k] * Wl[k * 128 + c];
        acc += agg[n * 7 + k] * iv * Wr[k * 128 + c];
    }
    float sc = g[c] * rsqrtf(rv[c] + BN_EPS);
    float v  = (acc - rm[c]) * sc + be[c];
    out[(long long)n * 128 + c] = fmaxf(v, 0.0f);
}

// ---------------------------------------------------------------- 128-wide edge scatter
__global__ __launch_bounds__(256)
void scatter128_k(const int* __restrict__ src, const int* __restrict__ dst,
                  const float* __restrict__ h, float* __restrict__ agg) {
    long long gid = (long long)blockIdx.x * 256 + threadIdx.x;
    int e = (int)(gid >> 7);
    int k = (int)(gid & 127);
    if (e < NE) {
        atomicAdd(&agg[(long long)dst[e] * 128 + k],
                  h[(long long)src[e] * 128 + k]);
    }
}

// ---------------------------------------------------------------- WMMA fp32 SAGE GEMM
// out[N,DOUT] = relu( BN( X@Wl + bl + (AGG*inv_row)@Wr ) ), din fixed at 128.
// Block = 8 waves = 8 M-tiles sharing one 16-column N-tile; B staged in LDS in
// WMMA-fragment order so the hot loop is ds_load_b64 + global_load_b64 + wmma.
template <int DOUT>
__global__ __launch_bounds__(256)
void sage_gemm_wmma_k(const float* __restrict__ X,   const float* __restrict__ AGG,
                      const float* __restrict__ INV,
                      const float* __restrict__ Wl,  const float* __restrict__ Wr,
                      const float* __restrict__ bl,  const float* __restrict__ g,
                      const float* __restrict__ be,  const float* __restrict__ rm,
                      const float* __restrict__ rv,
                      float* __restrict__ out) {
    constexpr int TILES_N = DOUT / 16;
    constexpr int TILES_M = NN / 16;          // 6250
    constexpr int WPB     = 8;                // waves (M-tiles) per block

    __shared__ float lWl[32 * 64];            // [k-step][lane*2 + j] fragment order
    __shared__ float lWr[32 * 64];

    const int tid     = threadIdx.x;
    const int tn      = blockIdx.x % TILES_N;
    const int tmb     = (blockIdx.x / TILES_N) * WPB;
    const int colBase = tn * 16;

    // ---- stage the 128x16 weight tiles into LDS in fragment order (once/block)
    for (int i = tid; i < 2048; i += 256) {
        int s  = i >> 6;                      // k-step (0..31)
        int L  = (i >> 1) & 31;               // lane
        int j  = i & 1;
        int kk = 4 * s + 2 * (L >> 4) + j;    // K index inside 128
        int c  = colBase + (L & 15);
        lWl[i] = Wl[kk * DOUT + c];
        lWr[i] = Wr[kk * DOUT + c];
    }
    __syncthreads();

    const int lane = tid & 31;
    const int wave = tid >> 5;
    int tm = tmb + wave;
    if (tm > TILES_M - 1) tm = TILES_M - 1;   // uniform per-wave clamp (dup rewrite)

    const int l16  = lane & 15;
    const int half = lane >> 4;               // A/B 32-bit layout: half picks K=0,1 vs 2,3
    const int arow = tm * 16 + l16;
    const int col  = colBase + l16;

    const float* ax = X   + (size_t)arow * 128 + half * 2;
    const float* ag = AGG + (size_t)arow * 128 + half * 2;
    const float* bL = lWl + lane * 2;
    const float* bR = lWr + lane * 2;

    v8f accA = {0.f, 0.f, 0.f, 0.f, 0.f, 0.f, 0.f, 0.f};   // X @ Wl
    v8f accB = {0.f, 0.f, 0.f, 0.f, 0.f, 0.f, 0.f, 0.f};   // AGG @ Wr

#pragma unroll
    for (int s = 0; s < 32; ++s) {
        v2f a0 = *(const v2f*)(ax + 4 * s);
        v2f b0 = *(const v2f*)(bL + 64 * s);
        accA = __builtin_amdgcn_wmma_f32_16x16x4_f32(false, a0, false, b0,
                                                     (short)0, accA, false, false);
        v2f a1 = *(const v2f*)(ag + 4 * s);
        v2f b1 = *(const v2f*)(bR + 64 * s);
        accB = __builtin_amdgcn_wmma_f32_16x16x4_f32(false, a1, false, b1,
                                                     (short)0, accB, false, false);
    }

    // epilogue: + bl, BatchNorm(eval), ReLU ; 1/deg row scale folded in here
    float sc  = g[col] * rsqrtf(rv[col] + BN_EPS);
    float off = (bl[col] - rm[col]) * sc + be[col];
#pragma unroll
    for (int r = 0; r < 8; ++r) {
        int orow = tm * 16 + r + half * 8;    // C/D layout: lanes>=16 hold M=8..15
        float v  = accA[r] + INV[orow] * accB[r];
        v = v * sc + off;
        out[(size_t)orow * DOUT + col] = fmaxf(v, 0.0f);
    }
}

// ---------------------------------------------------------------- global add pool
__global__ __launch_bounds__(256)
void pool_k(const float* __restrict__ xo, const int* __restrict__ batch,
            float* __restrict__ pooled) {
    long long gid = (long long)blockIdx.x * 256 + threadIdx.x;
    int n = (int)(gid >> 8);
    int c = (int)(gid & 255);
    if (n < NN)
        atomicAdd(&pooled[(long long)batch[n] * 256 + c],
                  xo[(long long)n * 256 + c]);
}

// ----------------------------------------------------------------
static inline size_t align256(size_t v) { return (v + 255) & ~(size_t)255; }

extern "C" void kernel_launch(void* const* d_in, const int* in_sizes, int n_in,
                              void* d_out, int out_size, void* d_ws, size_t ws_size,
                              hipStream_t stream) {
    (void)in_sizes; (void)n_in; (void)out_size; (void)ws_size;

    const float* x     = (const float*)d_in[0];
    const int*   ei    = (const int*)d_in[1];
    const int*   src   = ei;
    const int*   dst   = ei + NE;
    const int*   batch = (const int*)d_in[2];

    const float* P[21];
    for (int i = 0; i < 21; ++i) P[i] = (const float*)d_in[3 + i];
    // per layer i: P[7i+0]=Wl, +1=bl, +2=Wr, +3=g, +4=be, +5=rm, +6=rv

    // workspace layout
    char*  ws   = (char*)d_ws;
    size_t off  = 0;
    float* deg  = (float*)(ws + off); off += align256((size_t)NN * 4);          // deg -> inv_deg
    float* agg  = (float*)(ws + off); off += align256((size_t)NN * 128 * 4);    // reused per layer
    float* h1   = (float*)(ws + off); off += align256((size_t)NN * 128 * 4);
    float* h2   = (float*)(ws + off); off += align256((size_t)NN * 128 * 4);

    float* pooled = (float*)d_out;
    float* xout   = (float*)d_out + (size_t)NG * 256;

    constexpr int TILES_M  = NN / 16;                  // 6250
    constexpr int MBLOCKS  = (TILES_M + 7) / 8;        // 782 (last block clamps)

    // ---- degrees (shared by all layers)
    hipMemsetAsync(deg, 0, (size_t)NN * 4, stream);
    deg_count_k <<<(NE + 255) / 256, 256, 0, stream>>>(dst, deg);
    deg_invert_k<<<(NN + 255) / 256, 256, 0, stream>>>(deg);

    // ---- layer 0 : din=7, dout=128
    hipMemsetAsync(agg, 0, (size_t)NN * 7 * 4, stream);
    scatter7_k<<<(NE + 255) / 256, 256, 0, stream>>>(src, dst, x, agg);
    layer0_k<<<NN, 128, 0, stream>>>(x, agg, deg,
                                     P[0], P[1], P[2], P[3], P[4], P[5], P[6], h1);

    // ---- layer 1 : din=128, dout=128
    hipMemsetAsync(agg, 0, (size_t)NN * 128 * 4, stream);
    {
        long long thr = (long long)NE * 128;
        scatter128_k<<<(int)((thr + 255) / 256), 256, 0, stream>>>(src, dst, h1, agg);
        sage_gemm_wmma_k<128><<<MBLOCKS * (128 / 16), 256, 0, stream>>>(
            h1, agg, deg, P[7], P[9], P[8], P[10], P[11], P[12], P[13], h2);
    }

    // ---- layer 2 : din=128, dout=256
    hipMemsetAsync(agg, 0, (size_t)NN * 128 * 4, stream);
    {
        long long thr = (long long)NE * 128;
        scatter128_k<<<(int)((thr + 255) / 256), 256, 0, stream>>>(src, dst, h2, agg);
        sage_gemm_wmma_k<256><<<MBLOCKS * (256 / 16), 256, 0, stream>>>(
            h2, agg, deg, P[14], P[16], P[15], P[17], P[18], P[19], P[20], xout);
    }

    // ---- global add pool
    hipMemsetAsync(pooled, 0, (size_t)NG * 256 * 4, stream);
    {
        long long thr = (long long)NN * 256;
        pool_k<<<(int)((thr + 255) / 256), 256, 0, stream>>>(xout, batch, pooled);
    }
}